// InferenceLinearBucketTable_19129784336956
// MI455X (gfx1250) — compile-verified
//
#include <hip/hip_runtime.h>
#include <stdint.h>

#define BUCKET_CAP 128
#define WAVES_PER_BLOCK 8
#define THREADS (WAVES_PER_BLOCK * 32)
#define Q_PER_WAVE 8

typedef __attribute__((ext_vector_type(2))) long long ll2;

// ---------- CDNA5 async global->LDS helpers (gfx1250) ----------

// LDS byte offset from a generic pointer to a __shared__ object:
// flat addresses in the LDS aperture map to LDS via addr[31:0] (ISA 10.2).
__device__ __forceinline__ unsigned lds_off32(const void* p) {
    return (unsigned)(uintptr_t)p;
}

// One 16-byte async copy, global -> LDS, per active lane. Tracked by ASYNCcnt.
__device__ __forceinline__ void async_g2l_b128(unsigned lds_byte_off,
                                               const long long* gsrc) {
    asm volatile("global_load_async_to_lds_b128 %0, %1, off"
                 :
                 : "v"(lds_byte_off),
                   "v"((unsigned long long)(uintptr_t)gsrc)
                 : "memory");
}

template <int N>
__device__ __forceinline__ void wait_asynccnt() {
#if __has_builtin(__builtin_amdgcn_s_wait_asynccnt)
    __builtin_amdgcn_s_wait_asynccnt(N);
#else
    asm volatile("s_wait_asynccnt %0" :: "n"(N) : "memory");
#endif
}

__device__ __forceinline__ unsigned ballot32(bool p) {
#if __has_builtin(__builtin_amdgcn_ballot_w32)
    return __builtin_amdgcn_ballot_w32(p);
#else
    return (unsigned)__ballot(p);
#endif
}

// ---------- kernel ----------

__global__ __launch_bounds__(THREADS) void bucket_lookup_kernel(
    const long long* __restrict__ keys,
    const long long* __restrict__ table_ids,
    const long long* __restrict__ bucket_keys,
    const int*       __restrict__ bucket_sizes,
    const float*     __restrict__ bucket_scores,
    const long long* __restrict__ offsets,
    float* __restrict__ out_score,
    float* __restrict__ out_found,
    float* __restrict__ out_idx,
    int nq)
{
    // Per-wave double buffer: 8 waves * 2 * 128 * 8B = 16 KB (of 320 KB/WGP).
    __shared__ long long lds_keys[WAVES_PER_BLOCK][2][BUCKET_CAP];

    const int lane   = threadIdx.x & 31;
    const int w      = threadIdx.x >> 5;
    const int wid    = blockIdx.x * WAVES_PER_BLOCK + w;
    const int stride = gridDim.x * WAVES_PER_BLOCK;

    if (wid >= nq) return;  // wave-uniform

    // Stage query q into LDS buffer `parity`. Issues EXACTLY two async
    // instructions per wave (lane 0 forced active) so ASYNCcnt bookkeeping
    // is deterministic; lanes beyond the bucket's occupied size are EXEC-
    // masked off to cut HBM key traffic ~2x (invalid slots are masked in
    // the compare anyway, so stale LDS there is harmless).
    auto stage = [&](int q, int parity, long long& b_out, long long& k_out,
                     int& sz_out) {
        // Prefetch next iteration's scattered per-query loads (WGP scope:
        // consumed by this wave next iteration, so pull into near caches).
        if (q + stride < nq) {
            __builtin_prefetch(keys + q + stride, 0, 3);
            __builtin_prefetch(table_ids + q + stride, 0, 3);
        }
        const unsigned long long key = (unsigned long long)keys[q];
        const int tid = (int)table_ids[q];
        const unsigned long long start = (unsigned long long)offsets[tid];
        const unsigned long long end   = (unsigned long long)offsets[tid + 1];
        const unsigned long long nb    = end - start;
        // nb is a power of two for this dataset -> mask instead of 64-bit mod;
        // keep the generic remainder as an unlikely out-of-line fallback.
        unsigned long long r;
        if (__builtin_expect((nb & (nb - 1ull)) == 0ull, 1))
            r = key & (nb - 1ull);
        else
            r = key % nb;
        const long long b = (long long)(start + r);
        const int size = bucket_sizes[b];

        const int base = lane << 2;  // this lane's first slot (4 slots/lane)
        const long long* g = bucket_keys + b * (long long)BUCKET_CAP + base;
        const unsigned l0 = lds_off32(&lds_keys[w][parity][base]);

        if ((base < size) || (lane == 0))       async_g2l_b128(l0,        g);
        if ((base + 2 < size) || (lane == 0))   async_g2l_b128(l0 + 16u,  g + 2);

        b_out = b; k_out = (long long)key; sz_out = size;
    };

    // Compare buffer `parity` against the query; first match wins (lane
    // order == slot order, matching jnp.argmax over the match mask).
    auto process = [&](int parity, long long b, long long key, int size,
                       int q) {
        const long long* lk = &lds_keys[w][parity][0];
        const int base = lane << 2;
        // Two 16B vector reads -> ds_load_b128 each (32B-aligned per lane).
        const ll2 p0 = *(const ll2*)(lk + base);
        const ll2 p1 = *(const ll2*)(lk + base + 2);

        const bool m0 = (p0.x == key) & (base + 0 < size);
        const bool m1 = (p0.y == key) & (base + 1 < size);
        const bool m2 = (p1.x == key) & (base + 2 < size);
        const bool m3 = (p1.y == key) & (base + 3 < size);

        const int local = m0 ? 0 : (m1 ? 1 : (m2 ? 2 : (m3 ? 3 : 4)));
        const unsigned mask = ballot32(local < 4);

        float sc = 0.0f, fo = 0.0f, ix = -1.0f;
        if (mask) {  // wave-uniform branch
            const int fl    = __builtin_ctz(mask);
            const int lslot = __builtin_amdgcn_readlane(local, fl);
            const int slot  = (fl << 2) + lslot;
            const long long gidx = b * (long long)BUCKET_CAP + slot;
            fo = 1.0f;
            ix = (float)gidx;
            sc = bucket_scores[gidx];  // uniform address -> broadcast load
        }
        if (lane == 0) {
            out_score[q] = sc;
            out_found[q] = fo;
            out_idx[q]   = ix;
        }
    };

    // Software-pipelined grid-stride loop, pipeline depth 2:
    // while comparing bucket i from LDS, bucket i+1 is in flight via ASYNCcnt.
    long long cb, ck;
    int cs;
    int cq = wid;
    int parity = 0;
    stage(cq, parity, cb, ck, cs);

    for (int q = wid + stride;; q += stride) {
        const bool more = (q < nq);  // wave-uniform
        long long nb_ = 0, nk_ = 0;
        int ns_ = 0;
        if (more) {
            stage(q, parity ^ 1, nb_, nk_, ns_);
            wait_asynccnt<2>();  // <=2 outstanding => previous buffer landed
        } else {
            wait_asynccnt<0>();
        }
        process(parity, cb, ck, cs, cq);
        if (!more) break;
        cb = nb_; ck = nk_; cs = ns_; cq = q;
        parity ^= 1;
    }
}

// ---------- launcher ----------

extern "C" void kernel_launch(void* const* d_in, const int* in_sizes, int n_in,
                              void* d_out, int out_size, void* d_ws,
                              size_t ws_size, hipStream_t stream) {
    (void)n_in; (void)out_size; (void)d_ws; (void)ws_size;

    const long long* keys    = (const long long*)d_in[0];
    const long long* tids    = (const long long*)d_in[1];
    const long long* bkeys   = (const long long*)d_in[2];
    const int*       bsizes  = (const int*)d_in[3];
    const float*     bscores = (const float*)d_in[4];
    const long long* offs    = (const long long*)d_in[5];

    const int nq = in_sizes[0];
    float* out       = (float*)d_out;
    float* out_score = out;
    float* out_found = out + nq;
    float* out_idx   = out + 2 * (long long)nq;

    long long waves_needed = ((long long)nq + Q_PER_WAVE - 1) / Q_PER_WAVE;
    int blocks = (int)((waves_needed + WAVES_PER_BLOCK - 1) / WAVES_PER_BLOCK);
    if (blocks < 1) blocks = 1;

    bucket_lookup_kernel<<<blocks, THREADS, 0, stream>>>(
        keys, tids, bkeys, bsizes, bscores, offs,
        out_score, out_found, out_idx, nq);
}